// ImprovedYOLOLoss_38001870635760
// MI455X (gfx1250) — compile-verified
//
#include <hip/hip_runtime.h>
#include <stdint.h>

// ---------------- problem constants (from reference) ----------------
#define REG_MAX     16
#define NUM_CLASSES 30
#define NUM_CH      (4 * REG_MAX + NUM_CLASSES)   // 94
#define NB          128                            // batch
#define NM          50                             // targets per image
#define NA          2100                           // anchors
#define NA_PAD      2112                           // 132 chunks of 16
#define IMG_SIZE    320.0f
#define PAD_COORD   1.0e18f                        // sentinel: |a|^2 = 2e36 >> any real dist

typedef float v2f __attribute__((ext_vector_type(2)));
typedef float v8f __attribute__((ext_vector_type(8)));

// exact operand types for the async-to-LDS builtin (from clang diagnostic):
// param0: v4i in AS(1) (__device__), param1: v4i in AS(3) (__shared__)
typedef int v4i_gcc __attribute__((vector_size(4 * sizeof(int))));
typedef __attribute__((address_space(1))) v4i_gcc gbl_v4i;
typedef __attribute__((address_space(3))) v4i_gcc lds_v4i;

// ---------------- workspace layout ----------------
// [0, NB*NA) uint32   : per-(b,anchor) class bitmask (bit c set -> target_cls==1)
// then 4 floats       : acc[0]=cls_sum, acc[1]=box_sum, acc[2]=fg_count
#define WS_WORDS (NB * NA + 4)

__global__ void zero_ws_kernel(uint32_t* __restrict__ ws, int n) {
    int i = blockIdx.x * blockDim.x + threadIdx.x;
    if (i < n) ws[i] = 0u;
}

// ---------------- target->anchor assignment via WMMA f32 16x16x4 ----------------
// One wave (32 threads) per block handles 16 targets.
// dist^2(a, c) = |a|^2 - 2 a.c + |c|^2 ; |c|^2 constant per target row -> drop.
// A (16x4): row m = (cx_m, cy_m, 0, 0)  [lanes 0-15 hold K=0,1; lanes 16-31 hold K=2,3 = 0]
// B (4x16): col n = (-2 ax_n, -2 ay_n, 0, 0)
// C (16x16): every row = |a_n|^2
// D = A*B + C = per (target, anchor) squared-distance-minus-|c|^2.
__global__ __launch_bounds__(32) void assign_kernel(
    const float*   __restrict__ targets_xy,   // (NB, NM, 2)
    const int*     __restrict__ targets_cls,  // (NB, NM)
    const uint8_t* __restrict__ valid,        // (NB, NM) bool
    const float*   __restrict__ anchors,      // (NA, 2)
    uint32_t*      __restrict__ cls_mask)     // (NB, NA)
{
    const int lane   = threadIdx.x;     // 0..31
    const int col    = lane & 15;       // anchor column within tile
    const int hiHalf = lane >> 4;       // 0: rows 0-7 / K=0,1 ; 1: rows 8-15 / K=2,3
    const int base   = blockIdx.x * 16; // flat target index (b*NM + m) base

    // ---- stage anchors (padded) into LDS once per block ----
    __shared__ float2 sAnch[NA_PAD];                  // 16896 bytes
    {
        const int nvec4_real = (NA * 2) / 4;          // 1050 full float4s (exactly in range)
        const int nvec4_pad  = (NA_PAD * 2) / 4;      // 1056
        float4* sv = (float4*)sAnch;
#if defined(__has_builtin) && __has_builtin(__builtin_amdgcn_global_load_async_to_lds_b128)
        const float4* gsrc = (const float4*)anchors;
        for (int i = lane; i < nvec4_real; i += 32) {
            __builtin_amdgcn_global_load_async_to_lds_b128(
                (gbl_v4i*)(gsrc + i),
                (lds_v4i*)((char*)sv + (size_t)i * 16), 0, 0);
        }
        for (int i = nvec4_real + lane; i < nvec4_pad; i += 32) {
            sv[i] = make_float4(PAD_COORD, PAD_COORD, PAD_COORD, PAD_COORD);
        }
#if __has_builtin(__builtin_amdgcn_s_wait_asynccnt)
        __builtin_amdgcn_s_wait_asynccnt(0);
#else
        asm volatile("s_wait_asynccnt 0" ::: "memory");
#endif
#else
        const float4* gsrc = (const float4*)anchors;
        for (int i = lane; i < nvec4_pad; i += 32) {
            float4 v = (i < nvec4_real)
                           ? gsrc[i]
                           : make_float4(PAD_COORD, PAD_COORD, PAD_COORD, PAD_COORD);
            sv[i] = v;   // same-wave DS ordering guarantees visibility to later ds_load
        }
#endif
    }

    // ---- A matrix operand (2 VGPRs) ----
    v2f Amat;
    {
        int bm = base + col;            // only meaningful for lanes 0-15
        float cx = targets_xy[2 * bm + 0] * IMG_SIZE;
        float cy = targets_xy[2 * bm + 1] * IMG_SIZE;
        Amat[0] = hiHalf ? 0.0f : cx;   // K=0 (lanes 0-15) / K=2 (lanes 16-31, zero)
        Amat[1] = hiHalf ? 0.0f : cy;   // K=1 / K=3
    }

    float minv[8];
    int   mini[8];
#pragma unroll
    for (int j = 0; j < 8; ++j) { minv[j] = 3.4e38f; mini[j] = 0; }

    const int nchunks = NA_PAD / 16;    // 132
    for (int ch = 0; ch < nchunks; ++ch) {
        int n = ch * 16 + col;
        float2 a  = sAnch[n];           // ds_load_b64, branch-free
        float  cb = a.x * a.x + a.y * a.y;

        v2f Bmat;
        Bmat[0] = hiHalf ? 0.0f : (-2.0f * a.x);  // K=0 row / K=2 row (zero)
        Bmat[1] = hiHalf ? 0.0f : (-2.0f * a.y);  // K=1 row / K=3 row (zero)

        v8f C;
#pragma unroll
        for (int j = 0; j < 8; ++j) C[j] = cb;    // same bias for every row

        v8f D = __builtin_amdgcn_wmma_f32_16x16x4_f32(
            /*neg_a=*/false, Amat, /*neg_b=*/false, Bmat,
            /*c_mod=*/(short)0, C, /*reuse_a=*/false, /*reuse_b=*/false);

#pragma unroll
        for (int j = 0; j < 8; ++j) {
            float d = D[j];
            if (d < minv[j]) { minv[j] = d; mini[j] = n; }
        }
    }

    // min+argmin reduce across the 16 lanes of each half-wave
#pragma unroll
    for (int off = 1; off < 16; off <<= 1) {
#pragma unroll
        for (int j = 0; j < 8; ++j) {
            float ov = __shfl_xor(minv[j], off, 32);
            int   oi = __shfl_xor(mini[j], off, 32);
            if (ov < minv[j] || (ov == minv[j] && oi < mini[j])) {
                minv[j] = ov; mini[j] = oi;
            }
        }
    }

    // lane 0 holds rows 0-7, lane 16 holds rows 8-15
    if (col == 0) {
        int rowbase = hiHalf ? 8 : 0;
#pragma unroll
        for (int j = 0; j < 8; ++j) {
            int bm = base + rowbase + j;
            if (valid[bm]) {
                int b    = bm / NM;
                int c    = targets_cls[bm];
                int best = mini[j];
                atomicOr((unsigned int*)&cls_mask[b * NA + best], 1u << c);
            }
        }
    }
}

// ---------------- streaming loss pass ----------------
// loss_cls elem = softplus(x) - t*x ; t sparse via bitmask.
// loss_box: huber(mean_64(dist) - 1) summed over fg anchors (mask != 0).
__global__ __launch_bounds__(256) void loss_kernel(
    const float*    __restrict__ pred,      // (NB, 94, NA)
    const uint32_t* __restrict__ cls_mask,  // (NB, NA)
    float*          __restrict__ acc)       // cls_sum, box_sum, fg_cnt
{
    const int idx = blockIdx.x * blockDim.x + threadIdx.x;
    float clsSum = 0.0f, boxSum = 0.0f, cnt = 0.0f;

    if (idx < NB * NA) {
        const int b = idx / NA;
        const int a = idx - b * NA;
        const float* pb = pred + (size_t)b * NUM_CH * NA + a;
        const uint32_t mb = cls_mask[idx];

        // distribution channels 0..63: channel mean -> huber
        float s = 0.0f;
#pragma unroll
        for (int c = 0; c < 4 * REG_MAX; ++c) s += pb[(size_t)c * NA];
        float m  = s * (1.0f / 64.0f);
        float e  = m - 1.0f;
        float ae = fabsf(e);
        float hub = (ae <= 1.0f) ? 0.5f * e * e : (ae - 0.5f);
        if (mb != 0u) { boxSum = hub; cnt = 1.0f; }

        __builtin_prefetch(pb + (size_t)(4 * REG_MAX) * NA, 0, 0); // global_prefetch_b8

        // class channels 64..93
#pragma unroll
        for (int c = 0; c < NUM_CLASSES; ++c) {
            float x = pb[(size_t)(4 * REG_MAX + c) * NA];
            float sp = (x > 0.0f) ? (x + log1pf(expf(-x))) : log1pf(expf(x));
            clsSum += sp;
            if ((mb >> c) & 1u) clsSum -= x;
        }
    }

    // block reduction, then 3 float atomics per block
    __shared__ float sh[3 * 256];
    const int t = threadIdx.x;
    sh[t] = clsSum; sh[256 + t] = boxSum; sh[512 + t] = cnt;
    __syncthreads();
    for (int s = 128; s > 0; s >>= 1) {
        if (t < s) {
            sh[t]       += sh[t + s];
            sh[256 + t] += sh[256 + t + s];
            sh[512 + t] += sh[512 + t + s];
        }
        __syncthreads();
    }
    if (t == 0) {
        atomicAdd(&acc[0], sh[0]);
        atomicAdd(&acc[1], sh[256]);
        atomicAdd(&acc[2], sh[512]);
    }
}

__global__ void finalize_kernel(const float* __restrict__ acc,
                                float* __restrict__ out) {
    float cnt      = acc[2];
    float loss_box = (cnt > 0.0f) ? (acc[1] / fmaxf(cnt, 1.0f)) : 0.0f;
    float loss_cls = acc[0] / (float)((size_t)NB * NUM_CLASSES * NA);
    out[0] = loss_box;   // tuple order: (loss_box, loss_cls)
    out[1] = loss_cls;
}

// ---------------- launch ----------------
extern "C" void kernel_launch(void* const* d_in, const int* in_sizes, int n_in,
                              void* d_out, int out_size, void* d_ws, size_t ws_size,
                              hipStream_t stream) {
    const float*   pred        = (const float*)  d_in[0]; // (128, 94, 2100)
    const int*     targets_cls = (const int*)    d_in[1]; // (128, 50)
    const float*   targets_xy  = (const float*)  d_in[2]; // (128, 50, 2)
    const uint8_t* valid       = (const uint8_t*)d_in[3]; // (128, 50) bool
    const float*   anchors     = (const float*)  d_in[4]; // (2100, 2)

    uint32_t* cls_mask = (uint32_t*)d_ws;
    float*    acc      = (float*)(cls_mask + NB * NA);

    // 1) clear mask + accumulators
    zero_ws_kernel<<<(WS_WORDS + 255) / 256, 256, 0, stream>>>((uint32_t*)d_ws, WS_WORDS);

    // 2) WMMA-based nearest-anchor assignment: 6400 targets / 16 per wave
    assign_kernel<<<(NB * NM) / 16, 32, 0, stream>>>(
        targets_xy, targets_cls, valid, anchors, cls_mask);

    // 3) single streaming pass over pred (101 MB -> HBM-bound)
    loss_kernel<<<(NB * NA + 255) / 256, 256, 0, stream>>>(pred, cls_mask, acc);

    // 4) finalize two scalars
    finalize_kernel<<<1, 1, 0, stream>>>(acc, (float*)d_out);
}